// NonLocalBlock_37409165148834
// MI455X (gfx1250) — compile-verified
//
#include <hip/hip_runtime.h>

// Non-local block (spatial self-attention), B=4, C=128, CI=64, N=4096.
// All GEMM work via V_WMMA_F32_16X16X4_F32 (fp32 to match the reference),
// flash-style online softmax to avoid the 256MB attention matrix.
//
// Round-3 changes: __launch_bounds__(32,1) gives the allocator the full
// VGPR budget (round 2 spilled A-operands to scratch, putting a
// scratch_load + s_wait_loadcnt 0x0 in front of inner-loop WMMAs); the
// B-operand gather in k_proj/k_wconv is split into two chunks (16/8 steps) so peak
// register liveness stays under budget while the load clauses still issue
// ahead of the WMMA stream.

#define B_   4
#define C_   128
#define CI_  64
#define N_   4096
#define BN_EPS 1e-5f

typedef float v2f __attribute__((ext_vector_type(2)));
typedef float v8f __attribute__((ext_vector_type(8)));

// D = A(16x4) * B(4x16) + C(16x16), fp32.
static __device__ __forceinline__ v8f wmma4(v2f a, v2f b, v8f c) {
  return __builtin_amdgcn_wmma_f32_16x16x4_f32(
      /*neg_a=*/false, a, /*neg_b=*/false, b,
      /*c_mod=*/(short)0, c, /*reuse_a=*/false, /*reuse_b=*/false);
}

// ---------------------------------------------------------------------------
// Kernel 0: zero the BN statistic accumulators (ws is poisoned by harness).
// ---------------------------------------------------------------------------
__global__ void k_zero(float* __restrict__ sums, float* __restrict__ sq) {
  int i = threadIdx.x;
  if (i < C_) { sums[i] = 0.f; sq[i] = 0.f; }
}

// ---------------------------------------------------------------------------
// Kernel 1: the three 1x1-conv projections as WMMA GEMMs.
//   theta, phi stored [B, CI, N] (C-layout stores are coalesced).
//   g stored transposed [B, N, CI] so it is the coalesced B-operand of PV.
// One wave per block; each wave owns a 16(out-ch) x 256(tokens) strip.
// ---------------------------------------------------------------------------
__global__ void __launch_bounds__(32, 1)
k_proj(const float* __restrict__ x,
       const float* __restrict__ gw, const float* __restrict__ gb,
       const float* __restrict__ tw, const float* __restrict__ tb,
       const float* __restrict__ pw, const float* __restrict__ pb,
       float* __restrict__ theta, float* __restrict__ phi,
       float* __restrict__ g_t) {
  int id = blockIdx.x;
  int nc = id & 15;  id >>= 4;          // token chunk (256 tokens each)
  int ot = id & 3;   id >>= 2;          // output-channel tile (CI/16 = 4)
  int pj = id % 3;   int b = id / 3;    // projection, batch

  const float* w;  const float* bias;
  if (pj == 0)      { w = tw; bias = tb; }
  else if (pj == 1) { w = pw; bias = pb; }
  else              { w = gw; bias = gb; }

  const int lane = threadIdx.x;
  const int m16  = lane & 15;
  const int hi   = lane >> 4;           // 0: K pair (4t,4t+1), 1: (4t+2,4t+3)
  const int o0   = ot * 16;

  // Preload A = weight tile rows [16 x 128], K = C = 128 -> 32 k-steps.
  v2f a[32];
#pragma unroll
  for (int t = 0; t < 32; ++t) {
    int k = 4 * t + 2 * hi;
    a[t].x = w[(o0 + m16) * C_ + k];
    a[t].y = w[(o0 + m16) * C_ + k + 1];
  }
  float bv[8];
#pragma unroll
  for (int r = 0; r < 8; ++r) bv[r] = bias[o0 + r + hi * 8];

  const float* xb = x + (size_t)b * C_ * N_;
#pragma unroll 1
  for (int nt = 0; nt < 16; ++nt) {
    const int col = (nc * 16 + nt) * 16 + m16;
    v8f acc = {};
    // Two 16-step chunks: each chunk's 32 loads issue as one clause with
    // incremental waits, while peak register liveness stays bounded.
#pragma unroll
    for (int half = 0; half < 2; ++half) {
      v2f bops[16];
#pragma unroll
      for (int t = 0; t < 16; ++t) {
        int k = 4 * (half * 16 + t) + 2 * hi;
        bops[t].x = xb[(size_t)k * N_ + col];
        bops[t].y = xb[(size_t)(k + 1) * N_ + col];
      }
#pragma unroll
      for (int t = 0; t < 16; ++t) acc = wmma4(a[half * 16 + t], bops[t], acc);
    }

    if (pj < 2) {
      float* out = (pj == 0 ? theta : phi) + (size_t)b * CI_ * N_;
#pragma unroll
      for (int r = 0; r < 8; ++r)
        out[(size_t)(o0 + r + hi * 8) * N_ + col] = acc[r] + bv[r];
    } else {
      float* out = g_t + (size_t)b * N_ * CI_;  // [token, channel] (one-time scatter)
#pragma unroll
      for (int r = 0; r < 8; ++r)
        out[(size_t)col * CI_ + (o0 + r + hi * 8)] = acc[r] + bv[r];
    }
  }
}

// ---------------------------------------------------------------------------
// Kernel 2: flash attention. One wave per (batch, 16-query block).
//   f-tile layout (C-matrix): vgpr r, lanes 0-15 -> query r, key=lane;
//                             lanes 16-31 -> query 8+r, key=lane-16.
//   Row reductions via shfl_xor masks 1..8 (stay within 16-lane halves),
//   so running max/sum and the y accumulators stay perfectly lane-aligned.
//   P tile goes through a 16x17 LDS buffer to become the A operand of PV.
//   phi operands are double-buffered across key tiles (prefetch overlaps
//   the softmax); g operands are issued before the softmax and consumed
//   after it. No barriers: single wave, DS ops are in-order.
// Output y stored [B, CI, N] for the W-conv's coalesced B operand.
// ---------------------------------------------------------------------------
__global__ void __launch_bounds__(32, 1)
k_attn(const float* __restrict__ theta,
       const float* __restrict__ phi,
       const float* __restrict__ g_t,
       float* __restrict__ y_cn) {
  __shared__ float pbuf[16 * 17];       // padded: conflict-free transpose
  const int qb = blockIdx.x & 255;
  const int b  = blockIdx.x >> 8;
  const int q0 = qb * 16;
  const int lane = threadIdx.x;
  const int m16  = lane & 15;
  const int hi   = lane >> 4;

  const float* th = theta + (size_t)b * CI_ * N_;
  const float* ph = phi   + (size_t)b * CI_ * N_;
  const float* gt = g_t   + (size_t)b * N_ * CI_;

  // Preload A = theta^T for this query block: K = CI = 64 -> 16 k-steps.
  v2f aq[16];
#pragma unroll
  for (int t = 0; t < 16; ++t) {
    int k = 4 * t + 2 * hi;
    aq[t].x = th[(size_t)k * N_ + q0 + m16];
    aq[t].y = th[(size_t)(k + 1) * N_ + q0 + m16];
  }

  v8f y[4] = {v8f{}, v8f{}, v8f{}, v8f{}};    // [16 q x 64 ch] accumulator
  float Mx[8], Ls[8];
#pragma unroll
  for (int r = 0; r < 8; ++r) { Mx[r] = -__builtin_inff(); Ls[r] = 0.f; }

  auto load_phi = [&](int kt, v2f* dst) {
    const int m0 = kt * 16;
#pragma unroll
    for (int t = 0; t < 16; ++t) {
      int k = 4 * t + 2 * hi;
      dst[t].x = ph[(size_t)k * N_ + m0 + m16];
      dst[t].y = ph[(size_t)(k + 1) * N_ + m0 + m16];
    }
  };

  auto step = [&](int kt, v2f* pb, v2f* pn) {
    const int m0 = kt * 16;
    // ---- QK^T: f[16q x 16k] from prefetched phi operands ----
    v8f f = {};
#pragma unroll
    for (int t = 0; t < 16; ++t) f = wmma4(aq[t], pb[t], f);

    // ---- issue this tile's g operands (consumed after softmax) ----
    v2f gbv[16];
#pragma unroll
    for (int t = 0; t < 4; ++t) {
      int k = 4 * t + 2 * hi;
#pragma unroll
      for (int ct = 0; ct < 4; ++ct) {
        gbv[t * 4 + ct].x = gt[(size_t)(m0 + k) * CI_ + ct * 16 + m16];
        gbv[t * 4 + ct].y = gt[(size_t)(m0 + k + 1) * CI_ + ct * 16 + m16];
      }
    }
    // ---- prefetch next tile's phi operands; overlaps the softmax ----
    if (kt + 1 < 256) load_phi(kt + 1, pn);

    // ---- online softmax (per query row, replicated across its 16 lanes) ----
    float p[8];
#pragma unroll
    for (int r = 0; r < 8; ++r) {
      float v  = f[r];
      float mx = v;
      mx = fmaxf(mx, __shfl_xor(mx, 1, 32));
      mx = fmaxf(mx, __shfl_xor(mx, 2, 32));
      mx = fmaxf(mx, __shfl_xor(mx, 4, 32));
      mx = fmaxf(mx, __shfl_xor(mx, 8, 32));
      float newM  = fmaxf(Mx[r], mx);
      float scale = __expf(Mx[r] - newM);   // 0 on first tile (-inf)
      Mx[r] = newM;
      float pe = __expf(v - newM);
      p[r] = pe;
      float ls = pe;
      ls += __shfl_xor(ls, 1, 32);
      ls += __shfl_xor(ls, 2, 32);
      ls += __shfl_xor(ls, 4, 32);
      ls += __shfl_xor(ls, 8, 32);
      Ls[r] = Ls[r] * scale + ls;
      y[0][r] *= scale; y[1][r] *= scale; y[2][r] *= scale; y[3][r] *= scale;
    }
    // ---- transpose P through LDS: C-layout -> A-layout ----
    // Single wave: DS ops are in-order at LDS; compiler inserts dscnt waits.
#pragma unroll
    for (int r = 0; r < 8; ++r)
      pbuf[(r + hi * 8) * 17 + m16] = p[r];
    // ---- PV: y[16q x 64c] += P[16x16] * g[16k x 64c] ----
#pragma unroll
    for (int t = 0; t < 4; ++t) {
      int k = 4 * t + 2 * hi;
      v2f pa;
      pa.x = pbuf[m16 * 17 + k];
      pa.y = pbuf[m16 * 17 + k + 1];
#pragma unroll
      for (int ct = 0; ct < 4; ++ct)
        y[ct] = wmma4(pa, gbv[t * 4 + ct], y[ct]);
    }
  };

  v2f pbA[16], pbB[16];
  load_phi(0, pbA);
#pragma unroll 1
  for (int kt2 = 0; kt2 < 256; kt2 += 2) {  // explicit ping-pong, no indirection
    step(kt2,     pbA, pbB);
    step(kt2 + 1, pbB, pbA);
  }

  // normalize and store y as [CI, N] (one-time scatter; W-conv reads coalesced)
  float inv[8];
#pragma unroll
  for (int r = 0; r < 8; ++r) inv[r] = 1.f / Ls[r];
  float* yo = y_cn + (size_t)b * CI_ * N_;
#pragma unroll
  for (int ct = 0; ct < 4; ++ct)
#pragma unroll
    for (int r = 0; r < 8; ++r)
      yo[(size_t)(ct * 16 + m16) * N_ + (q0 + r + hi * 8)] = y[ct][r] * inv[r];
}

// ---------------------------------------------------------------------------
// Kernel 3: W 1x1-conv (C x CI GEMM) -> w_y written into d_out, plus
// per-channel sum / sum-of-squares accumulated for BatchNorm.
// ---------------------------------------------------------------------------
__global__ void __launch_bounds__(32, 1)
k_wconv(const float* __restrict__ y_cn,
        const float* __restrict__ ww, const float* __restrict__ wb,
        float* __restrict__ wy,
        float* __restrict__ sums, float* __restrict__ sq) {
  int id = blockIdx.x;
  int nc = id & 15;  id >>= 4;
  int ot = id & 7;   int b = id >> 3;
  const int lane = threadIdx.x;
  const int m16  = lane & 15;
  const int hi   = lane >> 4;
  const int o0   = ot * 16;

  v2f a[16];                              // K = CI = 64 -> 16 k-steps
#pragma unroll
  for (int t = 0; t < 16; ++t) {
    int k = 4 * t + 2 * hi;
    a[t].x = ww[(o0 + m16) * CI_ + k];
    a[t].y = ww[(o0 + m16) * CI_ + k + 1];
  }
  float bv[8];
#pragma unroll
  for (int r = 0; r < 8; ++r) bv[r] = wb[o0 + r + hi * 8];

  const float* yb  = y_cn + (size_t)b * CI_ * N_;
  float*       wyb = wy   + (size_t)b * C_ * N_;
  float s1[8] = {0,0,0,0,0,0,0,0}, s2[8] = {0,0,0,0,0,0,0,0};

#pragma unroll 1
  for (int nt = 0; nt < 16; ++nt) {
    const int col = (nc * 16 + nt) * 16 + m16;
    v8f acc = {};
#pragma unroll
    for (int half = 0; half < 2; ++half) {
      v2f bops[8];
#pragma unroll
      for (int t = 0; t < 8; ++t) {
        int k = 4 * (half * 8 + t) + 2 * hi;
        bops[t].x = yb[(size_t)k * N_ + col];
        bops[t].y = yb[(size_t)(k + 1) * N_ + col];
      }
#pragma unroll
      for (int t = 0; t < 8; ++t) acc = wmma4(a[half * 8 + t], bops[t], acc);
    }

#pragma unroll
    for (int r = 0; r < 8; ++r) {
      float v = acc[r] + bv[r];
      wyb[(size_t)(o0 + r + hi * 8) * N_ + col] = v;
      s1[r] += v;
      s2[r] += v * v;
    }
  }
  // reduce the per-lane column partial sums across each 16-lane half,
  // then one atomic per output channel from lane 0 / lane 16.
#pragma unroll
  for (int r = 0; r < 8; ++r) {
    float v = s1[r], w2 = s2[r];
    v  += __shfl_xor(v, 1, 32);  w2 += __shfl_xor(w2, 1, 32);
    v  += __shfl_xor(v, 2, 32);  w2 += __shfl_xor(w2, 2, 32);
    v  += __shfl_xor(v, 4, 32);  w2 += __shfl_xor(w2, 4, 32);
    v  += __shfl_xor(v, 8, 32);  w2 += __shfl_xor(w2, 8, 32);
    if (m16 == 0) {
      atomicAdd(&sums[o0 + r + hi * 8], v);
      atomicAdd(&sq  [o0 + r + hi * 8], w2);
    }
  }
}

// ---------------------------------------------------------------------------
// Kernel 4: finalize BN (training-mode batch stats) + residual, in place on
// d_out (which currently holds w_y).
// ---------------------------------------------------------------------------
__global__ void k_bn(const float* __restrict__ x,
                     const float* __restrict__ gamma, const float* __restrict__ beta,
                     const float* __restrict__ sums,  const float* __restrict__ sq,
                     float* __restrict__ out) {
  size_t i = (size_t)blockIdx.x * blockDim.x + threadIdx.x;
  if (i >= (size_t)B_ * C_ * N_) return;
  int o = (int)((i / N_) % C_);
  const float cnt  = (float)(B_ * N_);
  float mean = sums[o] / cnt;
  float var  = sq[o] / cnt - mean * mean;   // population var, matches jnp.var
  float v = out[i];
  out[i] = gamma[o] * (v - mean) * rsqrtf(var + BN_EPS) + beta[o] + x[i];
}

// ---------------------------------------------------------------------------
extern "C" void kernel_launch(void* const* d_in, const int* in_sizes, int n_in,
                              void* d_out, int out_size, void* d_ws, size_t ws_size,
                              hipStream_t stream) {
  const float* x     = (const float*)d_in[0];
  const float* g_w   = (const float*)d_in[1];
  const float* g_b   = (const float*)d_in[2];
  const float* th_w  = (const float*)d_in[3];
  const float* th_b  = (const float*)d_in[4];
  const float* ph_w  = (const float*)d_in[5];
  const float* ph_b  = (const float*)d_in[6];
  const float* w_w   = (const float*)d_in[7];
  const float* w_b   = (const float*)d_in[8];
  const float* gamma = (const float*)d_in[9];
  const float* beta  = (const float*)d_in[10];
  float* out = (float*)d_out;

  // Workspace: theta, phi (CI-major), g (token-major), y (CI-major), BN stats.
  // Total: 4 * B*CI*N floats + 2*C floats = 16 MB + 1 KB.
  float* ws    = (float*)d_ws;
  const size_t SZ = (size_t)B_ * CI_ * N_;
  float* theta = ws;
  float* phi   = theta + SZ;
  float* g_t   = phi + SZ;
  float* y_cn  = g_t + SZ;
  float* sums  = y_cn + SZ;
  float* sq    = sums + C_;

  k_zero <<<1, 128, 0, stream>>>(sums, sq);
  k_proj <<<B_ * 3 * 4 * 16, 32, 0, stream>>>(x, g_w, g_b, th_w, th_b, ph_w, ph_b,
                                              theta, phi, g_t);
  k_attn <<<B_ * (N_ / 16), 32, 0, stream>>>(theta, phi, g_t, y_cn);
  k_wconv<<<B_ * 8 * 16, 32, 0, stream>>>(y_cn, w_w, w_b, out, sums, sq);
  k_bn   <<<((size_t)B_ * C_ * N_ + 255) / 256, 256, 0, stream>>>(x, gamma, beta,
                                                                  sums, sq, out);
}